// ToRGBLayer_10677288697908
// MI455X (gfx1250) — compile-verified
//
#include <hip/hip_runtime.h>

#define W_DIM   512
#define IN_CH   256
#define OUT_CH  3
#define BATCH   8
#define HW      65536   // 256*256
#define CLAMP_V 256.0f

typedef float v2f __attribute__((ext_vector_type(2)));
typedef float v4f __attribute__((ext_vector_type(4)));
typedef float v8f __attribute__((ext_vector_type(8)));

// ---------------------------------------------------------------------------
// Kernel 1: styles GEMM (8x512 @ 512x256) on the f32 WMMA pipe, fused with
// affine bias, both normalization gains, and the o-channel conv weights.
// One workgroup, 16 waves; wave i owns output channels [16*i, 16*i+16).
// A (the w matrix) is staged in LDS padded to 16 rows so every lane can
// fetch its fragment unconditionally (no EXEC masking in the K loop).
// Emits mod_w[b][c][4] = {w0*style, w1*style, w2*style, 0} into d_ws.
// ---------------------------------------------------------------------------
__global__ __launch_bounds__(512) void modw_wmma_kernel(
    const float* __restrict__ w,        // [8][512]
    const float* __restrict__ affine_w, // [256][512]
    const float* __restrict__ affine_b, // [256]
    const float* __restrict__ weight,   // [3][256]
    float* __restrict__ modw)           // [8][256][4]
{
    __shared__ float wl[16 * W_DIM]; // 32 KB: w padded to 16 WMMA rows
    const int t = threadIdx.x;
    #pragma unroll
    for (int i = t; i < 16 * W_DIM; i += 512)
        wl[i] = (i < BATCH * W_DIM) ? w[i] : 0.0f;   // uniform per 512-chunk
    __syncthreads();

    const int wave  = t >> 5;          // 0..15 -> N tile
    const int lane  = t & 31;
    const int nlane = lane & 15;
    const int kk0   = (lane < 16) ? 0 : 2;   // K phase per half-wave (A and B)
    const float* arow = wl + (size_t)nlane * W_DIM + kk0;           // A row M=nlane
    const float* brow = affine_w + (size_t)(wave * 16 + nlane) * W_DIM + kk0;

    v8f acc = {0.f, 0.f, 0.f, 0.f, 0.f, 0.f, 0.f, 0.f};
    for (int k = 0; k < W_DIM; k += 4) {
        const v2f a  = *(const v2f*)(arow + k);   // {A[m][k+kk0], A[m][k+kk0+1]}
        const v2f bf = *(const v2f*)(brow + k);   // {B[k+kk0][n], B[k+kk0+1][n]}
        acc = __builtin_amdgcn_wmma_f32_16x16x4_f32(
            /*neg_a=*/false, a, /*neg_b=*/false, bf,
            /*c_mod=*/(short)0, acc, /*reuse_a=*/false, /*reuse_b=*/false);
    }

    // D layout: lanes 0-15, VGPR r -> element (M=r, N=lane). Batch rows 0..7
    // all live in lanes 0..15, so only those lanes write back.
    if (lane < 16) {
        const int n = wave * 16 + lane;
        const float bterm = affine_b[n];
        const float w0 = weight[0 * IN_CH + n];
        const float w1 = weight[1 * IN_CH + n];
        const float w2 = weight[2 * IN_CH + n];
        const float FC_GAIN = 0.044194173824159216f; // 1/sqrt(512)
        const float INV_SQ  = 0.0625f;               // 1/sqrt(256)
        #pragma unroll
        for (int r = 0; r < BATCH; ++r) {
            const float style = (acc[r] * FC_GAIN + bterm) * INV_SQ;
            v4f mw;
            mw.x = w0 * style; mw.y = w1 * style; mw.z = w2 * style; mw.w = 0.f;
            *(v4f*)(modw + (size_t)(r * IN_CH + n) * 4) = mw;
        }
    }
}

// ---------------------------------------------------------------------------
// Kernel 2: bandwidth-bound stream over x (512 MiB -> ~24 us roofline at
// 23.3 TB/s). Each thread owns 4 consecutive pixels (b128 loads), reduces
// over 256 channels with mod_w broadcast from LDS, then bias + clamp and
// non-temporal b128 stores. grid = (64, 8): 64 blocks x 256 threads x 4 px.
// ---------------------------------------------------------------------------
__device__ __forceinline__ v4f clamp4(v4f v) {
    v.x = fminf(fmaxf(v.x, -CLAMP_V), CLAMP_V);
    v.y = fminf(fmaxf(v.y, -CLAMP_V), CLAMP_V);
    v.z = fminf(fmaxf(v.z, -CLAMP_V), CLAMP_V);
    v.w = fminf(fmaxf(v.w, -CLAMP_V), CLAMP_V);
    return v;
}

__global__ __launch_bounds__(256) void torgb_stream_kernel(
    const float* __restrict__ x,     // [8][256][65536]
    const float* __restrict__ modw,  // [8][256][4]
    const float* __restrict__ bias,  // [3]
    float* __restrict__ out)         // [8][3][65536]
{
    __shared__ float smw[IN_CH * 4]; // 4 KB: mod_w for this batch, [c][4]
    const int t = threadIdx.x;
    const int b = blockIdx.y;
    ((v4f*)smw)[t] = ((const v4f*)(modw + (size_t)b * IN_CH * 4))[t];
    __syncthreads();

    const size_t p0 = ((size_t)blockIdx.x * 256 + t) * 4;
    const float* xb = x + (size_t)b * IN_CH * HW + p0;

    v4f a0 = {0.f, 0.f, 0.f, 0.f};
    v4f a1 = a0, a2 = a0;

    #pragma unroll 8
    for (int c = 0; c < IN_CH; ++c) {
        // x streamed exactly once: non-temporal to keep it out of L2 residency
        const v4f xv = __builtin_nontemporal_load((const v4f*)(xb + (size_t)c * HW));
        const float w0 = smw[c * 4 + 0];
        const float w1 = smw[c * 4 + 1];
        const float w2 = smw[c * 4 + 2];
        a0 += w0 * xv;
        a1 += w1 * xv;
        a2 += w2 * xv;
    }

    a0 = clamp4(a0 + bias[0]);
    a1 = clamp4(a1 + bias[1]);
    a2 = clamp4(a2 + bias[2]);

    float* ob = out + (size_t)b * OUT_CH * HW + p0;
    __builtin_nontemporal_store(a0, (v4f*)(ob + 0 * HW));
    __builtin_nontemporal_store(a1, (v4f*)(ob + 1 * HW));
    __builtin_nontemporal_store(a2, (v4f*)(ob + 2 * HW));
}

extern "C" void kernel_launch(void* const* d_in, const int* in_sizes, int n_in,
                              void* d_out, int out_size, void* d_ws, size_t ws_size,
                              hipStream_t stream) {
    const float* x        = (const float*)d_in[0];
    const float* w        = (const float*)d_in[1];
    const float* affine_w = (const float*)d_in[2];
    const float* affine_b = (const float*)d_in[3];
    const float* weight   = (const float*)d_in[4];
    const float* bias     = (const float*)d_in[5];
    float* out  = (float*)d_out;
    float* modw = (float*)d_ws;   // needs 8*256*4*4 = 32 KB scratch

    modw_wmma_kernel<<<dim3(1), dim3(512), 0, stream>>>(
        w, affine_w, affine_b, weight, modw);

    torgb_stream_kernel<<<dim3(HW / (256 * 4), BATCH), dim3(256), 0, stream>>>(
        x, modw, bias, out);
}